// DigitCaps_50285477101872
// MI455X (gfx1250) — compile-verified
//
#include <hip/hip_runtime.h>
#include <math.h>

typedef __attribute__((ext_vector_type(2))) float v2f;
typedef __attribute__((ext_vector_type(8))) float v8f;

#define B_N   128
#define R_N   4608
#define K_N   2
#define O_N   32
#define I_N   16
#define KO_N  64          // K_N * O_N
#define CHUNK 32          // r's per block
#define NBLK  (R_N / CHUNK)   // 144
#define SKO   (B_N * KO_N)    // 8192 = size of s/v

static_assert(R_N % CHUNK == 0, "chunking must divide R");

__device__ __forceinline__ v8f wmma_f32(v2f a, v2f b, v8f c) {
  // D = A(16x4) * B(4x16) + C(16x16), fp32
  return __builtin_amdgcn_wmma_f32_16x16x4_f32(
      /*neg_a=*/false, a, /*neg_b=*/false, b,
      /*c_mod=*/(short)0, c, /*reuse_a=*/false, /*reuse_b=*/false);
}

// ---------------------------------------------------------------------------
// init: c = 1/R (softmax of zeros over axis r), b_ij = 0
// ---------------------------------------------------------------------------
__global__ void init_kernel(float* __restrict__ c, float* __restrict__ bij) {
  int i = blockIdx.x * blockDim.x + threadIdx.x;
  if (i < R_N * K_N) {
    c[i]   = 1.0f / (float)R_N;
    bij[i] = 0.0f;
  }
}

// ---------------------------------------------------------------------------
// s_partial: per block (32 r's), accumulate sum_r c[r,k] * u_hat[b,r,ko]
// into a [128 x 64] tile via WMMA; write deterministic per-block partial.
// 8 waves: wave w owns b-rows [16w, 16w+16), all 4 ko-tiles.
// ---------------------------------------------------------------------------
__global__ __launch_bounds__(256) void s_partial_kernel(
    const float* __restrict__ x, const float* __restrict__ W,
    const float* __restrict__ c, float* __restrict__ partial) {
  const int rc   = blockIdx.x;
  const int wave = threadIdx.x >> 5;
  const int lane = threadIdx.x & 31;
  const int m    = lane & 15;       // M (b) / N (ko) index within tile
  const int hi   = lane >> 4;       // upper half of wave
  const int kb   = hi << 1;         // K-pair base: 0 or 2
  const int b0   = wave << 4;

  v8f acc[4] = {};

  const int rbeg = rc * CHUNK;
  for (int rr = 0; rr < CHUNK; ++rr) {
    const int r = rbeg + rr;
    // A fragments: x[b0+m, r, i0+kb], i0 = 0,4,8,12
    const float* xp = x + ((size_t)(b0 + m) * R_N + r) * I_N + kb;
    v2f a0 = *(const v2f*)(xp + 0);
    v2f a1 = *(const v2f*)(xp + 4);
    v2f a2 = *(const v2f*)(xp + 8);
    v2f a3 = *(const v2f*)(xp + 12);
    const float cw0 = c[2 * r + 0];
    const float cw1 = c[2 * r + 1];
    // W[r, ko, i] with flat index r*1024 + ko*16 + i  (ko = k*32+o)
    const float* wp = W + (size_t)r * (K_N * O_N * I_N) + kb;
#pragma unroll
    for (int t = 0; t < 4; ++t) {
      const float cw = (t < 2) ? cw0 : cw1;   // k = (t*16+n) >> 5 = t>>1
      const float* wq = wp + (t * 16 + m) * I_N;
      v2f bf0 = *(const v2f*)(wq + 0)  * cw;
      v2f bf1 = *(const v2f*)(wq + 4)  * cw;
      v2f bf2 = *(const v2f*)(wq + 8)  * cw;
      v2f bf3 = *(const v2f*)(wq + 12) * cw;
      acc[t] = wmma_f32(a0, bf0, acc[t]);
      acc[t] = wmma_f32(a1, bf1, acc[t]);
      acc[t] = wmma_f32(a2, bf2, acc[t]);
      acc[t] = wmma_f32(a3, bf3, acc[t]);
    }
  }

  float* pp = partial + (size_t)rc * SKO;
#pragma unroll
  for (int t = 0; t < 4; ++t)
#pragma unroll
    for (int vv = 0; vv < 8; ++vv) {
      const int b = b0 + vv + (hi << 3);      // D row = v + 8*hi
      pp[b * KO_N + t * 16 + m] = acc[t][vv];
    }
}

// ---------------------------------------------------------------------------
// s_reduce: s[idx] = sum over NBLK partials (fixed order -> deterministic)
// ---------------------------------------------------------------------------
__global__ void s_reduce_kernel(const float* __restrict__ partial,
                                float* __restrict__ s) {
  int idx = blockIdx.x * blockDim.x + threadIdx.x;   // 0..8191
  float acc = 0.0f;
  for (int blk = 0; blk < NBLK; ++blk)
    acc += partial[(size_t)blk * SKO + idx];
  s[idx] = acc;
}

// ---------------------------------------------------------------------------
// squash: v = sn*s / ((0.5+sn)*sqrt(sn)), sn = sum_o s^2 ; one thread per (b,k)
// ---------------------------------------------------------------------------
__global__ __launch_bounds__(256) void squash_kernel(
    const float* __restrict__ s, float* __restrict__ v) {
  const int j = threadIdx.x;                 // (b,k) pair, 256 of them
  const float* sp = s + j * O_N;
  float sn = 0.0f;
#pragma unroll
  for (int o = 0; o < O_N; ++o) { float t = sp[o]; sn += t * t; }
  const float scale = sn / ((0.5f + sn) * sqrtf(sn));
  float* vp = v + j * O_N;
#pragma unroll
  for (int o = 0; o < O_N; ++o) vp[o] = scale * sp[o];
}

// ---------------------------------------------------------------------------
// a_update: recompute u_hat per r via WMMA, a[r,k] = mean_b sum_o u_hat*v,
// bij[r,k] += a[r,k]. Each r owned by exactly one block -> deterministic.
// ---------------------------------------------------------------------------
__global__ __launch_bounds__(256) void a_update_kernel(
    const float* __restrict__ x, const float* __restrict__ W,
    const float* __restrict__ vj, float* __restrict__ bij) {
  __shared__ float red[8][2];
  const int rc   = blockIdx.x;
  const int wave = threadIdx.x >> 5;
  const int lane = threadIdx.x & 31;
  const int m    = lane & 15;
  const int hi   = lane >> 4;
  const int kb   = hi << 1;
  const int b0   = wave << 4;

  // v is r-invariant: preload this lane's 32 relevant values
  float vr[4][8];
#pragma unroll
  for (int t = 0; t < 4; ++t)
#pragma unroll
    for (int vv = 0; vv < 8; ++vv)
      vr[t][vv] = vj[(b0 + vv + (hi << 3)) * KO_N + t * 16 + m];

  const int rbeg = rc * CHUNK;
  for (int rr = 0; rr < CHUNK; ++rr) {
    const int r = rbeg + rr;
    const float* xp = x + ((size_t)(b0 + m) * R_N + r) * I_N + kb;
    v2f a0 = *(const v2f*)(xp + 0);
    v2f a1 = *(const v2f*)(xp + 4);
    v2f a2 = *(const v2f*)(xp + 8);
    v2f a3 = *(const v2f*)(xp + 12);
    const float* wp = W + (size_t)r * (K_N * O_N * I_N) + kb;

    v8f acc[4] = {};
#pragma unroll
    for (int t = 0; t < 4; ++t) {
      const float* wq = wp + (t * 16 + m) * I_N;
      v2f bf0 = *(const v2f*)(wq + 0);
      v2f bf1 = *(const v2f*)(wq + 4);
      v2f bf2 = *(const v2f*)(wq + 8);
      v2f bf3 = *(const v2f*)(wq + 12);
      acc[t] = wmma_f32(a0, bf0, acc[t]);
      acc[t] = wmma_f32(a1, bf1, acc[t]);
      acc[t] = wmma_f32(a2, bf2, acc[t]);
      acc[t] = wmma_f32(a3, bf3, acc[t]);
    }

    // per-lane partial dots, per k (tiles 0,1 -> k=0 ; tiles 2,3 -> k=1)
    float p0 = 0.0f, p1 = 0.0f;
#pragma unroll
    for (int vv = 0; vv < 8; ++vv) {
      p0 += acc[0][vv] * vr[0][vv] + acc[1][vv] * vr[1][vv];
      p1 += acc[2][vv] * vr[2][vv] + acc[3][vv] * vr[3][vv];
    }
    // wave32 reduction
#pragma unroll
    for (int off = 16; off >= 1; off >>= 1) {
      p0 += __shfl_xor(p0, off, 32);
      p1 += __shfl_xor(p1, off, 32);
    }
    if (lane == 0) { red[wave][0] = p0; red[wave][1] = p1; }
    __syncthreads();
    if (threadIdx.x < 2) {
      float tot = 0.0f;
#pragma unroll
      for (int w = 0; w < 8; ++w) tot += red[w][threadIdx.x];
      bij[2 * r + threadIdx.x] += tot * (1.0f / (float)B_N);
    }
    __syncthreads();
  }
}

// ---------------------------------------------------------------------------
// softmax over r (axis=1) for each k ; single block, LDS tree reductions
// ---------------------------------------------------------------------------
__global__ __launch_bounds__(256) void softmax_kernel(
    const float* __restrict__ bij, float* __restrict__ c) {
  __shared__ float lds[256];
  const int tid = threadIdx.x;
  for (int k = 0; k < K_N; ++k) {
    float mx = -3.402823e38f;
    for (int r = tid; r < R_N; r += 256) mx = fmaxf(mx, bij[2 * r + k]);
    lds[tid] = mx; __syncthreads();
    for (int sft = 128; sft > 0; sft >>= 1) {
      if (tid < sft) lds[tid] = fmaxf(lds[tid], lds[tid + sft]);
      __syncthreads();
    }
    mx = lds[0]; __syncthreads();

    float sum = 0.0f;
    for (int r = tid; r < R_N; r += 256) sum += expf(bij[2 * r + k] - mx);
    lds[tid] = sum; __syncthreads();
    for (int sft = 128; sft > 0; sft >>= 1) {
      if (tid < sft) lds[tid] += lds[tid + sft];
      __syncthreads();
    }
    const float inv = 1.0f / lds[0]; __syncthreads();

    for (int r = tid; r < R_N; r += 256)
      c[2 * r + k] = expf(bij[2 * r + k] - mx) * inv;
    __syncthreads();
  }
}

// ---------------------------------------------------------------------------
extern "C" void kernel_launch(void* const* d_in, const int* in_sizes, int n_in,
                              void* d_out, int out_size, void* d_ws, size_t ws_size,
                              hipStream_t stream) {
  (void)in_sizes; (void)n_in; (void)out_size; (void)ws_size;
  const float* x = (const float*)d_in[0];   // [B, R, I]
  const float* W = (const float*)d_in[1];   // [1, R, K, O, I]
  float* out = (float*)d_out;               // [B, K, O]

  float* c       = (float*)d_ws;            // R*K
  float* bij     = c + R_N * K_N;           // R*K
  float* s       = bij + R_N * K_N;         // B*K*O
  float* v       = s + SKO;                 // B*K*O
  float* partial = v + SKO;                 // NBLK * SKO  (~4.7 MB)

  init_kernel<<<(R_N * K_N + 255) / 256, 256, 0, stream>>>(c, bij);

  for (int it = 0; it < 3; ++it) {
    s_partial_kernel<<<NBLK, 256, 0, stream>>>(x, W, c, partial);
    s_reduce_kernel<<<SKO / 256, 256, 0, stream>>>(partial, s);
    squash_kernel<<<1, 256, 0, stream>>>(s, (it == 2) ? out : v);
    if (it < 2) {
      a_update_kernel<<<NBLK, 256, 0, stream>>>(x, W, v, bij);
      softmax_kernel<<<1, 256, 0, stream>>>(bij, c);
    }
  }
}